// MySoftmax_72730976190550
// MI455X (gfx1250) — compile-verified
//
#include <hip/hip_runtime.h>

// Problem constants from the reference: x is [B,H,S,S] = [2,12,1024,1024] fp32.
#define SDIM  1024     // row length (softmax axis)
#define NN    8        // MLP hidden width
#define TPB   256      // 8 wave32s per block
#define EPT   4        // elements per thread: 256*4 = 1024
#define NWAVE (TPB/32)

// Native clang vector (NOT HIP_vector_type) so nontemporal/128-bit ops accept it.
typedef float v4f __attribute__((ext_vector_type(4)));

// Fused: rowmax -> subtract -> per-element 1->8->1 MLP -> rowsum -> recip-MLP -> scale.
// Memory-bound (~201 MB traffic); one HBM read + one HBM write of x total.
// CDNA5 paths used: global_load_async_to_lds_b128 (ASYNCcnt), s_wait_asynccnt,
// wave32 shuffle reductions, non-temporal streaming store.
__global__ __launch_bounds__(TPB) void fused_learned_softmax(
    const float* __restrict__ x,
    const float* __restrict__ exp_w1, const float* __restrict__ exp_b1,
    const float* __restrict__ exp_w2, const float* __restrict__ exp_b2,
    const float* __restrict__ div_w1, const float* __restrict__ div_b1,
    const float* __restrict__ div_w2, const float* __restrict__ div_b2,
    float* __restrict__ out)
{
    __shared__ __align__(16) float row[SDIM];
    __shared__ float sredMax[NWAVE];
    __shared__ float sredSum[NWAVE];

    const int t = threadIdx.x;
    const long long rowBase = (long long)blockIdx.x * SDIM;
    const float* gsrc = x + rowBase + t * EPT;

    // Low 32 bits of the generic pointer to a __shared__ object are the
    // workgroup-relative LDS byte offset (ISA 10.2).
    unsigned ldsOff = (unsigned)(unsigned long long)(&row[t * EPT]);

    // CDNA5 async copy: 16 bytes/lane, global -> LDS, tracked by ASYNCcnt.
    asm volatile("global_load_async_to_lds_b128 %0, %1, off"
                 :: "v"(ldsOff), "v"(gsrc)
                 : "memory");

    // Overlap: fetch the 50 uniform weight scalars while the copy is in flight.
    // Uniform pointer + constant index -> scalar (s_load) path.
    float w1e[NN], b1e[NN], w2e[NN], w1d[NN], b1d[NN], w2d[NN];
    #pragma unroll
    for (int n = 0; n < NN; ++n) {
        w1e[n] = exp_w1[n]; b1e[n] = exp_b1[n]; w2e[n] = exp_w2[n];
        w1d[n] = div_w1[n]; b1d[n] = div_b1[n]; w2d[n] = div_w2[n];
    }
    const float c2e = exp_b2[0];
    const float c2d = div_b2[0];

    // Wait for this wave's async transfers, then make the row visible block-wide.
    asm volatile("s_wait_asynccnt 0x0" ::: "memory");
    __syncthreads();

    // One ds_load_b128 per lane.
    const v4f v = *(const v4f*)(&row[t * EPT]);

    // ---- Row max ----
    float pmax = fmaxf(fmaxf(v.x, v.y), fmaxf(v.z, v.w));
    #pragma unroll
    for (int off = 16; off >= 1; off >>= 1)
        pmax = fmaxf(pmax, __shfl_xor(pmax, off, 32));
    const int wave = t >> 5;
    if ((t & 31) == 0) sredMax[wave] = pmax;
    __syncthreads();
    float rowmax = sredMax[0];
    #pragma unroll
    for (int i = 1; i < NWAVE; ++i) rowmax = fmaxf(rowmax, sredMax[i]);

    // ---- Per-element exp-MLP (registers only) ----
    float in[EPT] = { v.x - rowmax, v.y - rowmax, v.z - rowmax, v.w - rowmax };
    float eo[EPT];
    float psum = 0.f;
    #pragma unroll
    for (int j = 0; j < EPT; ++j) {
        float acc = c2e;
        #pragma unroll
        for (int n = 0; n < NN; ++n) {
            float h = fmaxf(fmaf(in[j], w1e[n], b1e[n]), 0.f);
            acc = fmaf(h, w2e[n], acc);
        }
        eo[j] = acc;
        psum += acc;
    }

    // ---- Row sum ----
    #pragma unroll
    for (int off = 16; off >= 1; off >>= 1)
        psum += __shfl_xor(psum, off, 32);
    if ((t & 31) == 0) sredSum[wave] = psum;
    __syncthreads();
    float rowsum = sredSum[0];
    #pragma unroll
    for (int i = 1; i < NWAVE; ++i) rowsum += sredSum[i];

    // ---- Reciprocal-MLP on the scalar row sum (redundant per lane, 16 FLOPs) ----
    float recip = c2d;
    #pragma unroll
    for (int n = 0; n < NN; ++n) {
        float h = fmaxf(fmaf(rowsum, w1d[n], b1d[n]), 0.f);
        recip = fmaf(h, w2d[n], recip);
    }

    // ---- Scale + streaming store (output is written once, never re-read) ----
    v4f o;
    o.x = eo[0] * recip;
    o.y = eo[1] * recip;
    o.z = eo[2] * recip;
    o.w = eo[3] * recip;
    __builtin_nontemporal_store(o, (v4f*)(out + rowBase + t * EPT));
}

extern "C" void kernel_launch(void* const* d_in, const int* in_sizes, int n_in,
                              void* d_out, int out_size, void* d_ws, size_t ws_size,
                              hipStream_t stream) {
    const float* x      = (const float*)d_in[0];
    const float* exp_w1 = (const float*)d_in[1];
    const float* exp_b1 = (const float*)d_in[2];
    const float* exp_w2 = (const float*)d_in[3];
    const float* exp_b2 = (const float*)d_in[4];
    const float* div_w1 = (const float*)d_in[5];
    const float* div_b1 = (const float*)d_in[6];
    const float* div_w2 = (const float*)d_in[7];
    const float* div_b2 = (const float*)d_in[8];
    float* out = (float*)d_out;

    const int rows = in_sizes[0] / SDIM;   // B*H*S = 24576
    fused_learned_softmax<<<rows, TPB, 0, stream>>>(
        x, exp_w1, exp_b1, exp_w2, exp_b2,
        div_w1, div_b1, div_w2, div_b2, out);
}